// SimpleLogisticRegressionModel_33088428048486
// MI455X (gfx1250) — compile-verified
//
#include <hip/hip_runtime.h>
#include <stdint.h>

#define VOCAB 50257
#define NLAB  128

typedef __attribute__((ext_vector_type(4))) float floatx4;

__global__ __launch_bounds__(256) void gather_logits_kernel(
    const int*   __restrict__ ids,
    const float* __restrict__ W,      // [NLAB, VOCAB] row-major
    const float* __restrict__ bias,   // [NLAB]
    float*       __restrict__ out,    // [ntok, NLAB]
    int ntok)
{
    __shared__ __align__(16) float s_bias[NLAB];

    const int lane = threadIdx.x & 31;

    // ---- CDNA5 async global->LDS staging of the bias vector -------------
    // Wave 0 only: 32 lanes x b128 = 512 B = all 128 bias floats in one
    // async-tensor-path instruction (tracked on ASYNCcnt, not LOADcnt).
    if (threadIdx.x < 32) {
        unsigned lds_addr =
            (unsigned)(uintptr_t)(&s_bias[0]) + (unsigned)(lane * 16);
        const float* gsrc = bias + lane * 4;
        asm volatile("global_load_async_to_lds_b128 %0, %1, off"
                     :
                     : "v"(lds_addr), "v"(gsrc)
                     : "memory");
    }
    asm volatile("s_wait_asynccnt 0" ::: "memory");
    __syncthreads();

    // Each lane keeps its 4 bias values in registers across the token loop.
    const int l0 = lane * 4;
    floatx4 bv = *(const floatx4*)(&s_bias[l0]);

    // One wave per token; wave-uniform token index -> scalar load of the id.
    int nwaves = (int)((gridDim.x * blockDim.x) >> 5);
    int wid    = (int)((blockIdx.x * blockDim.x + threadIdx.x) >> 5);
    wid = __builtin_amdgcn_readfirstlane(wid);

    for (int t = wid; t < ntok; t += nwaves) {
        int id = ids[t];                 // uniform address -> s_load_b32
        const float* col = W + id;       // W[l][id] = col[l * VOCAB]

        floatx4 r;
        r.x = col[(l0 + 0) * VOCAB];
        r.y = col[(l0 + 1) * VOCAB];
        r.z = col[(l0 + 2) * VOCAB];
        r.w = col[(l0 + 3) * VOCAB];

        r.x += bv.x;
        r.y += bv.y;
        r.z += bv.z;
        r.w += bv.w;

        // Streaming 512 B/wave store; NT hint keeps W resident in L2.
        floatx4* dst = (floatx4*)(out + (size_t)t * NLAB + l0);
        __builtin_nontemporal_store(r, dst);
    }
}

extern "C" void kernel_launch(void* const* d_in, const int* in_sizes, int n_in,
                              void* d_out, int out_size, void* d_ws, size_t ws_size,
                              hipStream_t stream) {
    const int*   ids = (const int*)  d_in[0];   // [B*S] token ids
    const float* W   = (const float*)d_in[1];   // [NLAB, VOCAB]
    const float* b   = (const float*)d_in[2];   // [NLAB]
    float*       out = (float*)d_out;           // [B*S, NLAB]

    int ntok = in_sizes[0];                     // 64 * 2048 = 131072

    // 2048 blocks * 8 waves = 16384 waves -> 8 tokens per wave.
    dim3 grid(2048), block(256);
    gather_logits_kernel<<<grid, block, 0, stream>>>(ids, W, b, out, ntok);
}